// Process_new_61624190763140
// MI455X (gfx1250) — compile-verified
//
#include <hip/hip_runtime.h>

#define MCOUNT 20
#define RCOUNT 20
#define BATCH  400
#define STEPF  1e-4f
#define VBPAD  32      // padded metabolite stride in LDS (cols 20..31 are zeros)

typedef float v2f __attribute__((ext_vector_type(2)));
typedef float v8f __attribute__((ext_vector_type(8)));

// ---------------------------------------------------------------------------
// Kernel 1: mass-action simulation. One thread per sample, state in LDS.
// ---------------------------------------------------------------------------
__global__ void __launch_bounds__(128)
sim_kernel(const float* __restrict__ x,
           const float* __restrict__ sub,
           const float* __restrict__ prod,
           const int*   __restrict__ iters,
           float*       __restrict__ xout)
{
    // Per-reaction compact descriptors (uniform across the whole batch).
    __shared__ int   s_i0[RCOUNT];        // substrate index 0 (0..19)
    __shared__ int   s_i1[RCOUNT];        // substrate index 1, or 20 (dummy==1.0)
    __shared__ int   s_uidx[RCOUNT][4];   // update metabolite index, or 21 (trash)
    __shared__ float s_ucoef[RCOUNT][4];  // STEP * (prod - sub), or 0
    // Per-thread metabolite state: 20 metabolites + [20]=const 1.0 + [21]=trash.
    __shared__ float xs[128 * 22];

    const int tid = threadIdx.x;

    if (tid < RCOUNT) {
        const int r = tid;
        int i0 = 20, i1 = 20, cnt = 0, u = 0;
        int   uidx[4];
        float ucoef[4];
        #pragma unroll
        for (int m = 0; m < MCOUNT; ++m) {
            const float sv = sub[r * MCOUNT + m];
            const float pv = prod[r * MCOUNT + m];
            if (sv > 0.5f) { if (cnt == 0) i0 = m; else i1 = m; ++cnt; }
            const float d = pv - sv;            // exact: values are 0.0 / 1.0
            if (d != 0.0f && u < 4) { uidx[u] = m; ucoef[u] = STEPF * d; ++u; }
        }
        if (cnt < 2) i1 = 20;                   // single-substrate: rate = x[i0]*1.0
        for (; u < 4; ++u) { uidx[u] = 21; ucoef[u] = 0.0f; }
        s_i0[r] = i0; s_i1[r] = i1;
        #pragma unroll
        for (int k = 0; k < 4; ++k) { s_uidx[r][k] = uidx[k]; s_ucoef[r][k] = ucoef[k]; }
    }
    __syncthreads();

    const int b = blockIdx.x * blockDim.x + tid;
    if (b < BATCH) {
        float* px = &xs[tid * 22];
        #pragma unroll
        for (int m = 0; m < MCOUNT; ++m) px[m] = x[b * MCOUNT + m];
        px[20] = 1.0f;
        px[21] = 0.0f;

        const int n = iters[b];                 // mask (iters>i) == run n sweeps
        for (int it = 0; it < n; ++it) {
            #pragma unroll 4
            for (int r = 0; r < RCOUNT; ++r) {
                const float rate = px[s_i0[r]] * px[s_i1[r]];
                // update indices within a reaction are distinct -> sequential
                // RMW is equivalent to the reference's simultaneous update
                px[s_uidx[r][0]] += rate * s_ucoef[r][0];
                px[s_uidx[r][1]] += rate * s_ucoef[r][1];
                px[s_uidx[r][2]] += rate * s_ucoef[r][2];
                px[s_uidx[r][3]] += rate * s_ucoef[r][3];
            }
        }
        #pragma unroll
        for (int m = 0; m < MCOUNT; ++m) xout[b * MCOUNT + m] = px[m];
    }
}

// ---------------------------------------------------------------------------
// Kernel 2: correlation matrix via V_WMMA_F32_16X16X4_F32. vb is staged in LDS
// padded to 32 columns (cols 20..31 zero) so the 32x32 tiled product needs no
// EXEC-masked loads: zero padding comes from the data itself.
// ---------------------------------------------------------------------------
__global__ void __launch_bounds__(256)
corr_kernel(const float* __restrict__ xin, float* __restrict__ cout)
{
    __shared__ float s_mean[MCOUNT];
    __shared__ float s_inv[MCOUNT];
    __shared__ float vb[BATCH * VBPAD];   // 51.2 KB, row-major [b][32]

    const int tid = threadIdx.x;

    // Pass 1: column means.
    if (tid < MCOUNT) {
        float s = 0.0f;
        for (int b = 0; b < BATCH; ++b) s += xin[b * MCOUNT + tid];
        s_mean[tid] = s * (1.0f / (float)BATCH);
    }
    __syncthreads();

    // Stage vb = x - mean into padded LDS; pad columns get exact zeros.
    for (int i = tid; i < BATCH * VBPAD; i += 256) {
        const int m = i & (VBPAD - 1);
        const int b = i >> 5;
        vb[i] = (m < MCOUNT) ? (xin[b * MCOUNT + m] - s_mean[m]) : 0.0f;
    }
    __syncthreads();

    // Pass 2: inverse norms directly from vb (matches reference sum(vb*vb)).
    if (tid < MCOUNT) {
        float ss = 0.0f;
        for (int b = 0; b < BATCH; ++b) {
            const float v = vb[b * VBPAD + tid];
            ss += v * v;
        }
        s_inv[tid] = 1.0f / sqrtf(ss);
    }
    __syncthreads();

    // Wave 0 (lanes 0..31, EXEC all ones) computes vb^T * vb with WMMA.
    if (tid < 32) {
        const int lane    = tid;
        const int halfsel = lane >> 4;   // 0: lanes 0-15 hold K k+0/k+1; 1: K k+2/k+3
        const int l       = lane & 15;

        v8f acc00 = {}, acc01 = {}, acc10 = {}, acc11 = {};

        for (int k = 0; k < BATCH; k += 4) {
            const int k0 = k + halfsel * 2;
            // A(16x4)=vb^T and B(4x16)=vb share per-lane layout for matching
            // tiles (symmetric product): f0 = cols 0..15, f1 = cols 16..31.
            v2f f0, f1;
            f0.x = vb[(k0 + 0) * VBPAD + l];
            f0.y = vb[(k0 + 1) * VBPAD + l];
            f1.x = vb[(k0 + 0) * VBPAD + 16 + l];
            f1.y = vb[(k0 + 1) * VBPAD + 16 + l];

            acc00 = __builtin_amdgcn_wmma_f32_16x16x4_f32(false, f0, false, f0,
                                                          (short)0, acc00, false, false);
            acc01 = __builtin_amdgcn_wmma_f32_16x16x4_f32(false, f0, false, f1,
                                                          (short)0, acc01, false, false);
            acc10 = __builtin_amdgcn_wmma_f32_16x16x4_f32(false, f1, false, f0,
                                                          (short)0, acc10, false, false);
            acc11 = __builtin_amdgcn_wmma_f32_16x16x4_f32(false, f1, false, f1,
                                                          (short)0, acc11, false, false);
        }

        // D layout: VGPR r -> row r (lanes 0-15) / row r+8 (lanes 16-31), col = l.
        #pragma unroll
        for (int r = 0; r < 8; ++r) {
            const int rowBase = r + 8 * halfsel;
            const int col0 = l;
            if (rowBase < MCOUNT && col0 < MCOUNT)
                cout[rowBase * MCOUNT + col0] = acc00[r] * s_inv[rowBase] * s_inv[col0];
            if (rowBase < MCOUNT && (16 + col0) < MCOUNT)
                cout[rowBase * MCOUNT + 16 + col0] = acc01[r] * s_inv[rowBase] * s_inv[16 + col0];
            if ((16 + rowBase) < MCOUNT && col0 < MCOUNT)
                cout[(16 + rowBase) * MCOUNT + col0] = acc10[r] * s_inv[16 + rowBase] * s_inv[col0];
            if ((16 + rowBase) < MCOUNT && (16 + col0) < MCOUNT)
                cout[(16 + rowBase) * MCOUNT + 16 + col0] =
                    acc11[r] * s_inv[16 + rowBase] * s_inv[16 + col0];
        }
    }
}

// ---------------------------------------------------------------------------
extern "C" void kernel_launch(void* const* d_in, const int* in_sizes, int n_in,
                              void* d_out, int out_size, void* d_ws, size_t ws_size,
                              hipStream_t stream)
{
    const float* x    = (const float*)d_in[0];   // [400,20]
    const float* sub  = (const float*)d_in[1];   // [20,20]
    const float* prod = (const float*)d_in[2];   // [20,20]
    const int*   its  = (const int*)  d_in[3];   // [400]

    float* xout = (float*)d_out;                  // first 8000 floats
    float* cmat = (float*)d_out + BATCH * MCOUNT; // next 400 floats

    sim_kernel<<<dim3((BATCH + 127) / 128), dim3(128), 0, stream>>>(x, sub, prod, its, xout);
    corr_kernel<<<dim3(1), dim3(256), 0, stream>>>(xout, cmat);
}